// LengthRegulator_43576738185337
// MI455X (gfx1250) — compile-verified
//
#include <hip/hip_runtime.h>
#include <hip/hip_bf16.h>

// LengthRegulator: (B,L,H)=(32,256,384), T=MAX_WAVE=2048.
// out[0 .. B*H*T)          = transposed gathered hidden states (f32)
// out[B*H*T .. B*H*T+B*T)  = validity masks (f32)
//
// Pure data movement: ~100 MB stores, 12 MB reads (L2-resident). HBM-store
// bound; floor ~4.8 us @ 23.3 TB/s. Strategy: fused single kernel, per-block
// duration scan in LDS, async global->LDS staging of gathered rows
// (CDNA5 GLOBAL_LOAD_ASYNC_TO_LDS_B128, tracked by ASYNCcnt), then fully
// coalesced non-temporal t-major stores (keep the gather set in L2).

namespace {
constexpr int B_ = 32;
constexpr int L_ = 256;
constexpr int H_ = 384;
constexpr int T_ = 2048;
constexpr int TS_ = 64;              // t-frames per block
constexpr int TILES_ = T_ / TS_;     // 32 tiles per batch row
constexpr int HSTRIDE_ = 388;        // 384 + 4 pad: 16B-aligned rows, 2-way bank conflict max
}

#if __has_builtin(__builtin_amdgcn_global_load_async_to_lds_b128)
#define LR_HAVE_ASYNC_LDS 1
#endif

// The builtin's pointer params are int4-vector pointers in addrspace(1)/(3)
// (clang prints addrspace(1) as "__device__" in HIP diagnostics).
typedef int lr_v4i __attribute__((vector_size(16)));
typedef __attribute__((address_space(1))) lr_v4i* lr_gptr;
typedef __attribute__((address_space(3))) lr_v4i* lr_lptr;

__device__ __forceinline__ void lr_stage16(const float* g, float* l) {
#ifdef LR_HAVE_ASYNC_LDS
  // Per-lane async DMA: copies 16 bytes from the global address g to the LDS address l, tracked by ASYNCcnt.
  __builtin_amdgcn_global_load_async_to_lds_b128(
      (lr_gptr)(g), (lr_lptr)(l), 0 /*imm offset*/, 0 /*cpol*/);
#else
  *(float4*)l = *(const float4*)g;
#endif
}

__device__ __forceinline__ void lr_wait_async() {
#ifdef LR_HAVE_ASYNC_LDS
#if __has_builtin(__builtin_amdgcn_s_wait_asynccnt)
  __builtin_amdgcn_s_wait_asynccnt(0);
#else
  asm volatile("s_wait_asynccnt 0" ::: "memory");
#endif
#endif
}

__device__ __forceinline__ void lr_store_nt(float* p, float v) {
#if __has_builtin(__builtin_nontemporal_store)
  __builtin_nontemporal_store(v, p);
#else
  *p = v;
#endif
}

__global__ __launch_bounds__(256, 1)
void LengthRegulator_43576738185337_kernel(const float* __restrict__ enc,
                                           const int* __restrict__ dur,
                                           float* __restrict__ out) {
  __shared__ int   csum[L_];
  __shared__ int   s_idx[TS_];
  __shared__ float smem[TS_ * HSTRIDE_];   // ~97 KB staging buffer

  const int b   = blockIdx.x >> 5;         // TILES_ == 32
  const int t0  = (blockIdx.x & (TILES_ - 1)) * TS_;
  const int tid = threadIdx.x;             // 256 threads

  // ---- inclusive scan of durations for this batch row (Hillis-Steele) ----
  csum[tid] = dur[b * L_ + tid];
  __syncthreads();
#pragma unroll
  for (int off = 1; off < L_; off <<= 1) {
    const int v = csum[tid];
    const int a = (tid >= off) ? csum[tid - off] : 0;
    __syncthreads();
    csum[tid] = v + a;
    __syncthreads();
  }
  const int total = csum[L_ - 1];

  // ---- searchsorted(csum, t, side='right') for the 64 frames of this tile ----
  if (tid < TS_) {
    const int t = t0 + tid;
    int lo = 0, hi = L_;
    while (lo < hi) {                       // first i with csum[i] > t
      const int mid = (lo + hi) >> 1;
      if (csum[mid] <= t) lo = mid + 1; else hi = mid;
    }
    const bool valid = (t < total);
    s_idx[tid] = valid ? lo : -1;
    lr_store_nt(out + (size_t)B_ * H_ * T_ + (size_t)b * T_ + t,
                valid ? 1.0f : 0.0f);
  }
  __syncthreads();

  // ---- stage the 64 gathered rows into LDS (async DMA, b128 per lane) ----
  // 64 rows * 96 float4 = 6144 chunks -> 24 per thread.
  const float* encB = enc + (size_t)b * L_ * H_;
#pragma unroll
  for (int k = 0; k < 24; ++k) {
    const int j   = k * 256 + tid;
    const int row = j / 96;                 // which t-frame's row
    const int c4  = j - row * 96;           // float4 column within row
    const int idx = s_idx[row];
    const int src = (idx < 0) ? 0 : idx;    // invalid frames zeroed at write
    lr_stage16(encB + (size_t)src * H_ + c4 * 4,
               &smem[row * HSTRIDE_ + c4 * 4]);
  }
  lr_wait_async();
  __syncthreads();

  // ---- coalesced non-temporal t-major stores: out[b][h][t0+tsub] ----
  const int  tsub  = tid & (TS_ - 1);       // lanes of a wave share h, span t
  const int  hq    = tid >> 6;              // 0..3
  const bool valid = (s_idx[tsub] >= 0);
  float* outRow = out + ((size_t)b * H_) * T_ + t0 + tsub;
#pragma unroll 4
  for (int h = hq; h < H_; h += 4) {
    const float v = valid ? smem[tsub * HSTRIDE_ + h] : 0.0f;
    lr_store_nt(outRow + (size_t)h * T_, v);
  }
}

extern "C" void kernel_launch(void* const* d_in, const int* in_sizes, int n_in,
                              void* d_out, int out_size, void* d_ws, size_t ws_size,
                              hipStream_t stream) {
  (void)in_sizes; (void)n_in; (void)d_ws; (void)ws_size; (void)out_size;
  const float* enc = (const float*)d_in[0];   // (32,256,384) f32
  const int*   dur = (const int*)d_in[1];     // (32,256) i32
  float*       out = (float*)d_out;           // 32*384*2048 + 32*2048 f32

  dim3 grid(B_ * TILES_);                     // 1024 blocks
  dim3 block(256);
  LengthRegulator_43576738185337_kernel<<<grid, block, 0, stream>>>(enc, dur, out);
}